// N2V_GCN_44023414784044
// MI455X (gfx1250) — compile-verified
//
#include <hip/hip_runtime.h>
#include <hip/hip_bf16.h>

typedef float v2f __attribute__((ext_vector_type(2)));
typedef float v8f __attribute__((ext_vector_type(8)));

__device__ inline v8f wmma_f32x4(v2f a, v2f b, v8f c) {
  // D = A(16x4) * B(4x16) + C  — V_WMMA_F32_16X16X4_F32
  return __builtin_amdgcn_wmma_f32_16x16x4_f32(
      /*neg_a=*/false, a, /*neg_b=*/false, b,
      /*c_mod=*/(short)0, c, /*reuse_a=*/false, /*reuse_b=*/false);
}

// ---------------------------------------------------------------- utilities
__global__ void zero_f32(float* __restrict__ p, int n) {
  int i = blockIdx.x * blockDim.x + threadIdx.x;
  if (i < n) p[i] = 0.0f;
}

// deg = segment_sum(ones, dst) over edges + self loops
__global__ void degree_kernel(const int* __restrict__ dst, float* __restrict__ deg,
                              int E, int N) {
  int t = blockIdx.x * blockDim.x + threadIdx.x;
  if (t < E) {
    atomicAdd(&deg[dst[t]], 1.0f);
  } else if (t < E + N) {
    atomicAdd(&deg[t - E], 1.0f);  // self loop
  }
}

// in-place: deg -> (deg>0 ? rsqrt(deg) : 0)
__global__ void dinv_kernel(float* __restrict__ deg, int N) {
  int i = blockIdx.x * blockDim.x + threadIdx.x;
  if (i < N) {
    float d = deg[i];
    deg[i] = (d > 0.0f) ? rsqrtf(d) : 0.0f;
  }
}

// ---------------------------------------------------- fused LayerNorm + x@W1
// one wave (32 threads) per 16-row tile; K=128 -> 32 WMMA f32 16x16x4 steps
__global__ __launch_bounds__(32)
void ln_xw1_kernel(const float* __restrict__ x, const float* __restrict__ ln_g,
                   const float* __restrict__ ln_b, const float* __restrict__ W1,
                   float* __restrict__ xw, int N) {
  __shared__ __attribute__((aligned(16))) float sx[16 * 128];  // x tile
  __shared__ float sw[128 * 16];                               // W1 row-major
  __shared__ float sg[128], sb[128];

  const int lane = threadIdx.x;
  const int tile = blockIdx.x;

  for (int i = lane; i < 128 * 16; i += 32) sw[i] = W1[i];
  for (int i = lane; i < 128; i += 32) { sg[i] = ln_g[i]; sb[i] = ln_b[i]; }

  // load 16x128 tile as float4, clamped at N
  for (int i = lane; i < 512; i += 32) {  // 16*128/4
    int row = tile * 16 + (i >> 5);
    if (row > N - 1) row = N - 1;
    ((float4*)sx)[i] = ((const float4*)x)[(size_t)row * 32 + (i & 31)];
  }
  __syncthreads();

  // per-row LN stats; both half-waves compute row (lane&15) -> no broadcast needed
  const int row = lane & 15;
  float s = 0.0f, s2 = 0.0f;
  for (int k = 0; k < 128; ++k) {
    float v = sx[row * 128 + k];
    s += v; s2 += v * v;
  }
  const float mu  = s * (1.0f / 128.0f);
  const float var = s2 * (1.0f / 128.0f) - mu * mu;
  const float rs  = rsqrtf(var + 1e-5f);

  const int kb  = (lane >> 4) * 2;  // k sub-offset for this half-wave
  const int col = lane & 15;
  v8f c = {};
  for (int kk = 0; kk < 32; ++kk) {
    int k0 = kk * 4 + kb;
    v2f a, b;
    float x0 = sx[row * 128 + k0];
    float x1 = sx[row * 128 + k0 + 1];
    a.x = (x0 - mu) * rs * sg[k0]     + sb[k0];
    a.y = (x1 - mu) * rs * sg[k0 + 1] + sb[k0 + 1];
    b.x = sw[k0 * 16 + col];
    b.y = sw[(k0 + 1) * 16 + col];
    c = wmma_f32x4(a, b, c);
  }
  // D tile: VGPR r -> M = r + 8*(lane>>4), N = lane&15
  const int mbase = tile * 16 + (lane >> 4) * 8;
  for (int r = 0; r < 8; ++r) {
    int m = mbase + r;
    if (m < N) xw[(size_t)m * 16 + col] = c[r];
  }
}

// ------------------------------------------------------------ edge scatter
// one thread per edge (or self-loop): acc[dst] += xw[src] * dinv[src]*dinv[dst]
__global__ void scatter_kernel(const float* __restrict__ xw, const int* __restrict__ src,
                               const int* __restrict__ dst, const float* __restrict__ dinv,
                               float* __restrict__ acc, int E, int N) {
  int t = blockIdx.x * blockDim.x + threadIdx.x;
  int s, d;
  if (t < E) { s = src[t]; d = dst[t]; }
  else if (t < E + N) { s = t - E; d = s; }
  else return;
  const float nm = dinv[s] * dinv[d];
  const float* ps = xw + (size_t)s * 16;
  float* pd = acc + (size_t)d * 16;
#pragma unroll
  for (int j = 0; j < 16; ++j) atomicAdd(&pd[j], ps[j] * nm);
}

__global__ void bias_relu_kernel(const float* __restrict__ acc, const float* __restrict__ b,
                                 float* __restrict__ h, int n16) {
  int i = blockIdx.x * blockDim.x + threadIdx.x;
  if (i < n16) h[i] = fmaxf(acc[i] + b[i & 15], 0.0f);
}

// ------------------------------------------------------------------ h1 @ W2
// one wave per 16-row tile; K=16 -> 4 WMMA steps; operands straight from L2
__global__ __launch_bounds__(32)
void hw2_kernel(const float* __restrict__ h, const float* __restrict__ W2,
                float* __restrict__ out, int N) {
  const int lane = threadIdx.x;
  const int tile = blockIdx.x;
  const int row  = lane & 15;
  const int kb   = (lane >> 4) * 2;
  const int col  = row;
  int rowg = tile * 16 + row;
  if (rowg > N - 1) rowg = N - 1;
  const float* hr = h + (size_t)rowg * 16;

  v8f c = {};
  for (int kk = 0; kk < 4; ++kk) {
    int k0 = kk * 4 + kb;
    v2f a, b;
    a.x = hr[k0];
    a.y = hr[k0 + 1];
    b.x = W2[k0 * 16 + col];
    b.y = W2[(k0 + 1) * 16 + col];
    c = wmma_f32x4(a, b, c);
  }
  const int mbase = tile * 16 + (lane >> 4) * 8;
  for (int r = 0; r < 8; ++r) {
    int m = mbase + r;
    if (m < N) out[(size_t)m * 16 + col] = c[r];
  }
}

// ------------------------------------------------- fused pair MLP (the hot loop)
// one wave per 16 pairs: gather u,v from L2-resident h2, build feat[16x64] in LDS,
// LN, feat@W3 via 16 WMMA K-steps x 2 N-tiles, ReLU, dot with W4 via shfl reduce.
__global__ __launch_bounds__(32)
void pair_kernel(const float* __restrict__ h2, const int* __restrict__ pu,
                 const int* __restrict__ pv, const float* __restrict__ eg,
                 const float* __restrict__ ebt, const float* __restrict__ W3,
                 const float* __restrict__ b3, const float* __restrict__ W4,
                 const float* __restrict__ b4, float* __restrict__ out, int P) {
  __shared__ float sf[16 * 64];    // feat tile
  __shared__ float sw3[64 * 32];   // W3 row-major
  __shared__ float sb3[32], sw4[32], sgm[64], sbt[64];

  const int lane = threadIdx.x;
  const int tile = blockIdx.x;

  for (int i = lane; i < 64 * 32; i += 32) sw3[i] = W3[i];
  if (lane < 32) { sb3[lane] = b3[lane]; sw4[lane] = W4[lane]; }
  for (int i = lane; i < 64; i += 32) { sgm[i] = eg[i]; sbt[i] = ebt[i]; }

  float mu = 0.0f, rs = 0.0f;
  if (lane < 16) {
    int p = tile * 16 + lane;
    if (p > P - 1) p = P - 1;
    const float* U = h2 + (size_t)pu[p] * 16;
    const float* V = h2 + (size_t)pv[p] * 16;
    float s = 0.0f, s2 = 0.0f;
#pragma unroll
    for (int j = 0; j < 16; ++j) {
      float u = U[j], v = V[j];
      float f0 = u, f1 = v, f2 = fabsf(u - v), f3 = u * v;
      sf[lane * 64 + j]      = f0;
      sf[lane * 64 + 16 + j] = f1;
      sf[lane * 64 + 32 + j] = f2;
      sf[lane * 64 + 48 + j] = f3;
      s  += f0 + f1 + f2 + f3;
      s2 += f0 * f0 + f1 * f1 + f2 * f2 + f3 * f3;
    }
    mu = s * (1.0f / 64.0f);
    float var = s2 * (1.0f / 64.0f) - mu * mu;
    rs = rsqrtf(var + 1e-5f);
  }
  __syncthreads();
  mu = __shfl(mu, lane & 15, 32);
  rs = __shfl(rs, lane & 15, 32);

  const int row = lane & 15;
  const int kb  = (lane >> 4) * 2;
  const int col = row;
  v8f c0 = {}, c1 = {};
  for (int kk = 0; kk < 16; ++kk) {
    int k0 = kk * 4 + kb;
    v2f a, bA, bB;
    float f0 = sf[row * 64 + k0];
    float f1 = sf[row * 64 + k0 + 1];
    a.x = (f0 - mu) * rs * sgm[k0]     + sbt[k0];
    a.y = (f1 - mu) * rs * sgm[k0 + 1] + sbt[k0 + 1];
    bA.x = sw3[k0 * 32 + col];        bA.y = sw3[(k0 + 1) * 32 + col];
    bB.x = sw3[k0 * 32 + 16 + col];   bB.y = sw3[(k0 + 1) * 32 + 16 + col];
    c0 = wmma_f32x4(a, bA, c0);
    c1 = wmma_f32x4(a, bB, c1);
  }

  const float w4a = sw4[col], w4b = sw4[16 + col];
  const float bb  = b4[0];
  const int mrow  = tile * 16 + (lane >> 4) * 8;
  for (int r = 0; r < 8; ++r) {
    float ha = fmaxf(c0[r] + sb3[col], 0.0f);
    float hb = fmaxf(c1[r] + sb3[16 + col], 0.0f);
    float t = ha * w4a + hb * w4b;     // partial over this lane's two hidden cols
    t += __shfl_xor(t, 1, 32);
    t += __shfl_xor(t, 2, 32);
    t += __shfl_xor(t, 4, 32);
    t += __shfl_xor(t, 8, 32);         // sum over the 16 lanes of this half-wave
    if ((lane & 15) == 0 && (mrow + r) < P) out[mrow + r] = t + bb;
  }
}

// ------------------------------------------------------------------ launcher
extern "C" void kernel_launch(void* const* d_in, const int* in_sizes, int n_in,
                              void* d_out, int out_size, void* d_ws, size_t ws_size,
                              hipStream_t stream) {
  const float* x    = (const float*)d_in[0];
  const int*   ei   = (const int*)d_in[1];
  const int*   ep   = (const int*)d_in[2];
  const float* ln_g = (const float*)d_in[3];
  const float* ln_b = (const float*)d_in[4];
  const float* W1   = (const float*)d_in[5];
  const float* b1   = (const float*)d_in[6];
  const float* W2   = (const float*)d_in[7];
  const float* b2   = (const float*)d_in[8];
  const float* eg   = (const float*)d_in[9];
  const float* ebt  = (const float*)d_in[10];
  const float* W3   = (const float*)d_in[11];
  const float* b3   = (const float*)d_in[12];
  const float* W4   = (const float*)d_in[13];
  const float* b4   = (const float*)d_in[14];

  const int N = in_sizes[0] / 128;
  const int E = in_sizes[1] / 2;
  const int P = in_sizes[2] / 2;

  const int* src = ei;          // edge_index[0]
  const int* dst = ei + E;      // edge_index[1]
  const int* pu  = ep;          // edge_pairs[0]
  const int* pv  = ep + P;      // edge_pairs[1]

  float* deg  = (float*)d_ws;                    // N
  float* bufA = deg  + N;                        // N*16
  float* bufB = bufA + (size_t)N * 16;           // N*16
  float* bufC = bufB + (size_t)N * 16;           // N*16

  const int n16 = N * 16;
  const int nodeTiles = (N + 15) / 16;
  const int pairTiles = (P + 15) / 16;

  // 1) degrees -> dinv
  zero_f32<<<(N + 255) / 256, 256, 0, stream>>>(deg, N);
  degree_kernel<<<(E + N + 255) / 256, 256, 0, stream>>>(dst, deg, E, N);
  dinv_kernel<<<(N + 255) / 256, 256, 0, stream>>>(deg, N);

  // 2) xw1 = LN(x) @ W1
  ln_xw1_kernel<<<nodeTiles, 32, 0, stream>>>(x, ln_g, ln_b, W1, bufA, N);

  // 3) conv1 aggregate + bias + relu -> h1 (h1 overwrites xw in bufA)
  zero_f32<<<(n16 + 255) / 256, 256, 0, stream>>>(bufB, n16);
  scatter_kernel<<<(E + N + 255) / 256, 256, 0, stream>>>(bufA, src, dst, deg, bufB, E, N);
  bias_relu_kernel<<<(n16 + 255) / 256, 256, 0, stream>>>(bufB, b1, bufA, n16);  // h1 -> bufA

  // 4) xw2 = h1 @ W2
  hw2_kernel<<<nodeTiles, 32, 0, stream>>>(bufA, W2, bufC, N);

  // 5) conv2 aggregate + bias + relu -> h2 (bufC)
  zero_f32<<<(n16 + 255) / 256, 256, 0, stream>>>(bufB, n16);
  scatter_kernel<<<(E + N + 255) / 256, 256, 0, stream>>>(bufC, src, dst, deg, bufB, E, N);
  bias_relu_kernel<<<(n16 + 255) / 256, 256, 0, stream>>>(bufB, b2, bufC, n16);  // h2 -> bufC

  // 6) pair MLP -> logits
  pair_kernel<<<pairTiles, 32, 0, stream>>>(bufC, pu, pv, eg, ebt, W3, b3, W4, b4,
                                            (float*)d_out, P);
}